// SpectralConv2d_83253646065781
// MI455X (gfx1250) — compile-verified
//
#include <hip/hip_runtime.h>
#include <hip/hip_bf16.h>
#include <math.h>

// SpectralConv2d on MI455X (gfx1250).
// Memory-bound problem (~140MB HBM traffic @ 23.3TB/s ~= 6-7us; only ~2 GFLOP),
// so all transform GEMMs use full-precision V_WMMA_F32_16X16X4_F32, and the
// forward kernel stages its strided 64x64 input tile into LDS with the Tensor
// Data Mover (TENSOR_LOAD_TO_LDS + s_wait_tensorcnt) instead of VGPR loads.
//
// Workspace layout (bytes):        size
//   avg   [16][8]  f32             512
//   nm    [8][16]  i32             512
//   Hht   [8][16][32][16][16] f32  4 MB    (forward cas-cas coefficients)
//   Ye    [16][32][32][16][16]f32  16 MB   (even Hartley weights, per n)
//   Yo    [16][32][32][16][16]f32  16 MB   (odd Hartley weights, per n)
//   Zht   [8][16][32][16][16] f32  4 MB    (spectral output, zero-padded to 16x16)
// total ~40 MB.

#define B_   8
#define CIN  32
#define COUT 32
#define S_   256
#define TWO_PI 6.28318530717958647692f
#define CAS64_STEP 0.09817477042468103871f   // 2*pi/64

typedef float v2f __attribute__((ext_vector_type(2)));
typedef float v8f __attribute__((ext_vector_type(8)));
typedef unsigned int u32x4 __attribute__((ext_vector_type(4)));
typedef int i32x4 __attribute__((ext_vector_type(4)));
typedef int i32x8 __attribute__((ext_vector_type(8)));

// D(16x16,f32) = A(16x4,f32) x B(4x16,f32) + C
__device__ __forceinline__ v8f wmma4(v2f a, v2f b, v8f c) {
  return __builtin_amdgcn_wmma_f32_16x16x4_f32(false, a, false, b, (short)0, c,
                                               false, false);
}

// TDM: load a 2D tile (rows x cols f32, row stride in elements) from global to LDS.
__device__ __forceinline__ void tdm_load_tile_f32(unsigned lds_byte_addr,
                                                  const void* gptr,
                                                  unsigned rows, unsigned cols,
                                                  unsigned stride_elems) {
  unsigned long long ga = (unsigned long long)(uintptr_t)gptr;
  u32x4 g0;
  g0[0] = 1u;                                   // count=1 (one valid descriptor)
  g0[1] = lds_byte_addr;                        // lds_addr [63:32]
  g0[2] = (unsigned)(ga & 0xffffffffu);         // global_addr lo
  g0[3] = (unsigned)((ga >> 32) & 0x01ffffffu)  // global_addr [56:32]
        | (2u << 30);                           // type=2 ("image")
  i32x8 g1;
  g1[0] = 0x00020000;                           // data_size=2 (4B); wg_mask=0
  g1[1] = (int)(cols << 16);                    // tensor_dim0[15:0] -> bits[79:64... (48-split)]
  g1[2] = (int)(rows << 16);                    // tensor_dim0 hi=0 | tensor_dim1 lo
  g1[3] = (int)(cols << 16);                    // tensor_dim1 hi=0 | tile_dim0
  g1[4] = (int)rows;                            // tile_dim1 | tile_dim2=0
  g1[5] = (int)stride_elems;                    // tensor_dim0_stride lo
  g1[6] = 0;                                    // stride hi | tensor_dim1_stride lo
  g1[7] = 0;
  i32x4 z4 = {0, 0, 0, 0};
#if __clang_major__ >= 23
  i32x8 z8 = {0, 0, 0, 0, 0, 0, 0, 0};
  __builtin_amdgcn_tensor_load_to_lds(g0, g1, z4, z4, z8, 0);
#else
  __builtin_amdgcn_tensor_load_to_lds(g0, g1, z4, z4, 0);
#endif
}

// ---------------------------------------------------------------- K0a: region means
__global__ void k_region_avg(const float* __restrict__ err, float* __restrict__ avg) {
  int b  = blockIdx.x >> 4;
  int rg = blockIdx.x & 15;
  int s1 = (rg >> 2) * 64, s2 = (rg & 3) * 64;
  const float* p = err + ((size_t)b * S_ + s1) * S_ + s2;
  float sum = 0.f;
  for (int t = threadIdx.x; t < 4096; t += 256) {
    int r = t >> 6, c = t & 63;
    sum += p[(size_t)r * S_ + c];
  }
  __shared__ float red[256];
  red[threadIdx.x] = sum;
  __syncthreads();
  for (int w = 128; w > 0; w >>= 1) {
    if (threadIdx.x < w) red[threadIdx.x] += red[threadIdx.x + w];
    __syncthreads();
  }
  if (threadIdx.x == 0) avg[rg * 8 + b] = red[0] * (1.f / 4096.f);
}

// ---------------------------------------------------------------- K0b: mode counts
__global__ void k_modes(const float* __restrict__ avg, int* __restrict__ nm) {
  int rg = threadIdx.x;
  if (rg < 16) {
    float mn = 1e30f, mx = -1e30f;
    for (int b = 0; b < B_; b++) {
      float v = avg[rg * 8 + b];
      mn = fminf(mn, v); mx = fmaxf(mx, v);
    }
    float d = mx - mn;
    for (int b = 0; b < B_; b++) {
      float norm = (d > 1e-8f) ? (avg[rg * 8 + b] - mn) / d : 0.f;
      int m = (int)(norm * 15.f) + 1;
      nm[b * 16 + rg] = m > 16 ? 16 : m;
    }
  }
}

// ---------------------------------------------------------------- K1: Hartley weights per n
__global__ void k_weights_ht(const float* __restrict__ w, float* __restrict__ Ye,
                             float* __restrict__ Yo) {
  int n = (blockIdx.x & 15) + 1;
  int i = blockIdx.x >> 4;
  __shared__ float casn[16];
  if (threadIdx.x < (unsigned)n) {
    float s, c;
    sincosf(TWO_PI * (float)threadIdx.x / (float)n, &s, &c);
    casn[threadIdx.x] = c + s;
  }
  __syncthreads();
  int n2 = n * n;
  for (int t = threadIdx.x; t < COUT * n2; t += 256) {
    int o = t / n2, k = t - o * n2;
    int u = k / n, v = k - u * n;
    int kf = (n2 - k) % n2;               // 1D-periodic flip over flattened n*n
    int uf = kf / n, vf = kf - uf * n;
    const float* wp = w + ((size_t)i * COUT + o) * 256;   // weights (Cin,Cout,16,16)
    float y = 0.f, yf = 0.f;
    for (int h = 0; h < n; h++)
      for (int ww = 0; ww < n; ww++) {
        float xv = wp[h * 16 + ww];
        y  += xv * casn[(u  * h + v  * ww) % n];
        yf += xv * casn[(uf * h + vf * ww) % n];
      }
    size_t off = (((size_t)(n - 1) * CIN + i) * COUT + o) * 256 + (size_t)u * 16 + v;
    Ye[off] = 0.5f * (y + yf);
    Yo[off] = 0.5f * (y - yf);
  }
}

// ---------------------------------------------------------------- K2: forward H = C*X*C^T (WMMA + TDM)
__global__ void __launch_bounds__(32) k_fwd(const float* __restrict__ x,
                                            float* __restrict__ Hht) {
  __shared__ __align__(16) float Xs[64 * 64];   // 16 KB, filled by TDM
  __shared__ __align__(16) float Ts[64 * 16];   // 4 KB
  __shared__ __align__(16) float cas64[64];
  int bx = blockIdx.x;
  int ci = bx & 31;
  int rg = (bx >> 5) & 15;
  int b  = bx >> 9;
  int s1 = (rg >> 2) * 64, s2 = (rg & 3) * 64;
  int lane = threadIdx.x;

  // Issue the tensor DMA first so it overlaps the cas-table build.
  const float* xp = x + (((size_t)(b * CIN + ci) * S_ + s1) * S_ + s2);
  tdm_load_tile_f32((unsigned)(uintptr_t)(&Xs[0]), xp, 64, 64, S_);

  {
    float s, c;
    sincosf(CAS64_STEP * (float)lane, &s, &c);        cas64[lane]      = c + s;
    sincosf(CAS64_STEP * (float)(lane + 32), &s, &c); cas64[lane + 32] = c + s;
  }
  __builtin_amdgcn_s_wait_tensorcnt(0);
  __syncthreads();

  int l15 = lane & 15, lhi = lane >> 4;
  // step 1: T(64x16) = X(64x64) * C^T(64x16), K accumulated 4 at a time
  for (int m = 0; m < 4; m++) {
    v8f acc = {};
    for (int k = 0; k < 16; k++) {
      int c0 = 4 * k + 2 * lhi;                       // K split: lanes<16 -> K=0,1 ; lanes>=16 -> K=2,3
      v2f a = *(const v2f*)&Xs[(m * 16 + l15) * 64 + c0];
      v2f bf; bf.x = cas64[(l15 * c0) & 63];
              bf.y = cas64[(l15 * (c0 + 1)) & 63];
      acc = wmma4(a, bf, acc);
    }
    for (int jj = 0; jj < 8; jj++)                    // C/D layout: VGPR j -> M=j (lo lanes), M=j+8 (hi)
      Ts[(m * 16 + jj + 8 * lhi) * 16 + l15] = acc[jj];
  }
  __syncthreads();
  // step 2: H(16x16) = C(16x64) * T(64x16)
  v8f acc = {};
  for (int k = 0; k < 16; k++) {
    int c0 = 4 * k + 2 * lhi;
    v2f a;  a.x = cas64[(l15 * c0) & 63];
            a.y = cas64[(l15 * (c0 + 1)) & 63];
    v2f bf; bf.x = Ts[c0 * 16 + l15];
            bf.y = Ts[(c0 + 1) * 16 + l15];
    acc = wmma4(a, bf, acc);
  }
  float* hp = Hht + ((size_t)(b * 16 + rg) * CIN + ci) * 256;
  for (int jj = 0; jj < 8; jj++)
    hp[(jj + 8 * lhi) * 16 + l15] = acc[jj];
}

// ---------------------------------------------------------------- K3: spectral channel mix
__global__ void k_mid(const float* __restrict__ Hht, const float* __restrict__ Ye,
                      const float* __restrict__ Yo, const int* __restrict__ nm,
                      float* __restrict__ Zht) {
  int b = blockIdx.x >> 4, rg = blockIdx.x & 15;
  int n = nm[b * 16 + rg];
  int n2 = n * n;
  __shared__ float Xs[CIN * 256];   // 32 KB : X flattened over n*n
  __shared__ float Zs[16 * 256];    // 16 KB : Z for 16 output channels at a time
  __shared__ float casn[16];
  if (threadIdx.x < (unsigned)n) {
    float s, c;
    sincosf(TWO_PI * (float)threadIdx.x / (float)n, &s, &c);
    casn[threadIdx.x] = c + s;
  }
  __syncthreads();
  const float* hbase = Hht + ((size_t)(b * 16 + rg) * CIN) * 256;
  // phase A: X = dht2(H[:, :n, :n])  (true 2D DHT: cas(2*pi*(u*h+v*w)/n))
  for (int t = threadIdx.x; t < CIN * n2; t += 256) {
    int i = t / n2, k = t - i * n2;
    int u = k / n, v = k - u * n;
    const float* hp = hbase + (size_t)i * 256;
    float acc = 0.f;
    for (int h = 0; h < n; h++)
      for (int w = 0; w < n; w++)
        acc += hp[h * 16 + w] * casn[(u * h + v * w) % n];
    Xs[i * 256 + k] = acc;
  }
  __syncthreads();
  size_t yoff = ((size_t)(n - 1) * CIN) * COUT * 256;
  float* zout = Zht + ((size_t)(b * 16 + rg) * COUT) * 256;
  for (int half = 0; half < 2; half++) {
    // phase B: Z[o][k] = sum_i X[i][k]*Ye + Xflip[i][k]*Yo
    for (int t = threadIdx.x; t < 16 * n2; t += 256) {
      int ol = t / n2, k = t - ol * n2;
      int o = half * 16 + ol;
      int u = k / n, v = k - u * n;
      int kf = (n2 - k) % n2;
      float acc = 0.f;
      for (int i = 0; i < CIN; i++) {
        size_t w_off = yoff + ((size_t)i * COUT + o) * 256 + (size_t)u * 16 + v;
        acc += Xs[i * 256 + k]  * Ye[w_off];
        acc += Xs[i * 256 + kf] * Yo[w_off];
      }
      Zs[ol * 256 + k] = acc;
    }
    __syncthreads();
    // phase C: blk = dht2(Z)/n2, written zero-padded into the 16x16 tile
    for (int t = threadIdx.x; t < 16 * 256; t += 256) {
      int ol = t >> 8, k = t & 255;
      int u = k >> 4, v = k & 15;
      float val = 0.f;
      if (u < n && v < n) {
        for (int h = 0; h < n; h++)
          for (int w = 0; w < n; w++)
            val += Zs[ol * 256 + h * n + w] * casn[(u * h + v * w) % n];
        val *= 1.f / (float)n2;
      }
      zout[(size_t)(half * 16 + ol) * 256 + k] = val;
    }
    __syncthreads();
  }
}

// ---------------------------------------------------------------- K4: inverse Out = C^T*Z*C /4096 (WMMA)
__global__ void __launch_bounds__(32) k_inv(const float* __restrict__ Zht,
                                            float* __restrict__ out) {
  __shared__ __align__(16) float Zs[256];
  __shared__ __align__(16) float Ts[64 * 16];
  __shared__ __align__(16) float cas64[64];
  int bx = blockIdx.x;
  int co = bx & 31, rg = (bx >> 5) & 15, b = bx >> 9;
  int s1 = (rg >> 2) * 64, s2 = (rg & 3) * 64;
  int lane = threadIdx.x;
  {
    float s, c;
    sincosf(CAS64_STEP * (float)lane, &s, &c);        cas64[lane]      = c + s;
    sincosf(CAS64_STEP * (float)(lane + 32), &s, &c); cas64[lane + 32] = c + s;
  }
  const float* zp = Zht + ((size_t)(b * 16 + rg) * COUT + co) * 256;
  for (int q = 0; q < 8; q++) Zs[q * 32 + lane] = zp[q * 32 + lane];
  __syncthreads();
  int l15 = lane & 15, lhi = lane >> 4;
  // step 1: T(64x16) = C^T rows of cas * Z(16x16), K=16
  for (int m = 0; m < 4; m++) {
    v8f acc = {};
    for (int k = 0; k < 4; k++) {
      int c0 = 4 * k + 2 * lhi;
      v2f a;  a.x = cas64[((m * 16 + l15) * c0) & 63];
              a.y = cas64[((m * 16 + l15) * (c0 + 1)) & 63];
      v2f bf; bf.x = Zs[c0 * 16 + l15];
              bf.y = Zs[(c0 + 1) * 16 + l15];
      acc = wmma4(a, bf, acc);
    }
    for (int jj = 0; jj < 8; jj++)
      Ts[(m * 16 + jj + 8 * lhi) * 16 + l15] = acc[jj];
  }
  __syncthreads();
  // step 2: Out(64x64) = T(64x16) * C(16x64), K=16, scaled by 1/4096
  float* op = out + (((size_t)(b * COUT + co) * S_ + s1) * S_ + s2);
  for (int m = 0; m < 4; m++) {
    for (int nt = 0; nt < 4; nt++) {
      v8f acc = {};
      for (int k = 0; k < 4; k++) {
        int c0 = 4 * k + 2 * lhi;
        v2f a = *(const v2f*)&Ts[(m * 16 + l15) * 16 + c0];
        int col = nt * 16 + l15;
        v2f bf; bf.x = cas64[(col * c0) & 63];
                bf.y = cas64[(col * (c0 + 1)) & 63];
        acc = wmma4(a, bf, acc);
      }
      for (int jj = 0; jj < 8; jj++) {
        int r = m * 16 + jj + 8 * lhi;
        op[(size_t)r * S_ + nt * 16 + l15] = acc[jj] * (1.f / 4096.f);
      }
    }
  }
}

// ---------------------------------------------------------------- launch
extern "C" void kernel_launch(void* const* d_in, const int* in_sizes, int n_in,
                              void* d_out, int out_size, void* d_ws, size_t ws_size,
                              hipStream_t stream) {
  const float* x   = (const float*)d_in[0];   // (8,32,256,256)
  const float* err = (const float*)d_in[1];   // (8,256,256)
  const float* w   = (const float*)d_in[2];   // (32,32,16,16)
  float* out = (float*)d_out;                 // (8,32,256,256)

  char* ws = (char*)d_ws;
  const size_t H_BYTES = (size_t)B_ * 16 * CIN * 256 * 4;      // 4 MB
  const size_t Y_BYTES = (size_t)16 * CIN * COUT * 256 * 4;    // 16 MB
  float* avg = (float*)ws;                                     // 512 B
  int*   nm  = (int*)(ws + 512);                               // 512 B
  float* Hht = (float*)(ws + 1024);
  float* Ye  = (float*)(ws + 1024 + H_BYTES);
  float* Yo  = (float*)(ws + 1024 + H_BYTES + Y_BYTES);
  float* Zht = (float*)(ws + 1024 + H_BYTES + 2 * Y_BYTES);
  (void)ws_size; (void)in_sizes; (void)n_in; (void)out_size;

  k_region_avg<<<128, 256, 0, stream>>>(err, avg);
  k_modes     <<<1,   32,  0, stream>>>(avg, nm);
  k_weights_ht<<<512, 256, 0, stream>>>(w, Ye, Yo);
  k_fwd       <<<B_ * 16 * CIN, 32, 0, stream>>>(x, Hht);
  k_mid       <<<B_ * 16, 256, 0, stream>>>(Hht, Ye, Yo, nm, Zht);
  k_inv       <<<B_ * 16 * COUT, 32, 0, stream>>>(Zht, out);
}